// GConvSE3Partial_32813550141517
// MI455X (gfx1250) — compile-verified
//
#include <hip/hip_runtime.h>

// SE(3) pairwise-conv partial: per-edge radial MLPs via fp32 WMMA (16x16x4),
// per-edge einsum on VALU. Wave32: one wave owns a tile of 16 edges.
// B-fragments are preloaded into explicit arrays and every layer runs two
// independent accumulator chains so loads clause up and WMMA latency overlaps.

typedef float v2f __attribute__((ext_vector_type(2)));
typedef float v8f __attribute__((ext_vector_type(8)));

#define E_TOT 160000
#define NCH 8
#define MIDW 32
#define EDIM 16
#define WAVES_PER_BLK 4

struct KParams {
  const float* feat0; const float* feat1; const float* w; const float* radial;
  const float* basis[4];
  const float* w1[4]; const float* b1[4];
  const float* w2[4]; const float* b2[4];
  const float* w3[4]; const float* b3[4];
  const int* src_idx;
  float* out;
};

__device__ __forceinline__ v8f wmma4(v2f a, v2f b, v8f c) {
  // D = A(16x4,f32) x B(4x16,f32) + C(16x16,f32)
  return __builtin_amdgcn_wmma_f32_16x16x4_f32(false, a, false, b,
                                               (short)0, c, false, false);
}

// B-fragment (rows k0..k0+3, cols n0..n0+15) of a KxN row-major weight matrix.
__device__ __forceinline__ v2f ldB(const float* __restrict__ W, int ncols,
                                   int k0, int n0, int m, int hi) {
  int k = k0 + 2 * hi;
  v2f b;
  b.x = W[(k    ) * ncols + n0 + m];
  b.y = W[(k + 1) * ncols + n0 + m];
  return b;
}

// Full radial MLP for one degree pair on a 16-edge tile.
// xa: A-fragments of ef[16 x 17(pad 20)]. Results R[16 x NF*64] -> sR.
template<int NF>
__device__ __forceinline__ void radial_mlp_tile(
    const float* __restrict__ w1, const float* __restrict__ bs1,
    const float* __restrict__ w2, const float* __restrict__ bs2,
    const float* __restrict__ w3, const float* __restrict__ bs3,
    const v2f* __restrict__ xa, float* __restrict__ sH, float* __restrict__ sR,
    int m, int hi) {
  // ---- layer 1: [16x17(pad20)] @ [17x32] + b1, ReLU -> sH[16][32]
  {
    v2f b0[5], b1f[5];
    #pragma unroll
    for (int c = 0; c < 4; ++c) {
      b0[c]  = ldB(w1, MIDW, 4 * c, 0,  m, hi);
      b1f[c] = ldB(w1, MIDW, 4 * c, 16, m, hi);
    }
    // K-chunk 4: only row 16 is real (rows 17..19 zero). Branchless select.
    {
      float wa = w1[16 * MIDW + m];
      float wb = w1[16 * MIDW + 16 + m];
      b0[4].x  = hi ? 0.f : wa;  b0[4].y  = 0.f;
      b1f[4].x = hi ? 0.f : wb;  b1f[4].y = 0.f;
    }
    float bias0 = bs1[m], bias1 = bs1[16 + m];
    v8f a0 = {}, a1 = {};
    #pragma unroll
    for (int c = 0; c < 5; ++c) {         // two independent WMMA chains
      a0 = wmma4(xa[c], b0[c],  a0);
      a1 = wmma4(xa[c], b1f[c], a1);
    }
    #pragma unroll
    for (int r = 0; r < 8; ++r) {
      float v0 = a0[r] + bias0, v1 = a1[r] + bias1;
      sH[(r + 8 * hi) * MIDW + m]      = v0 > 0.f ? v0 : 0.f;
      sH[(r + 8 * hi) * MIDW + 16 + m] = v1 > 0.f ? v1 : 0.f;
    }
  }
  // C-layout -> A-layout via LDS
  v2f ha[8];
  #pragma unroll
  for (int c = 0; c < 8; ++c) {
    ha[c].x = sH[m * MIDW + 4 * c + 2 * hi];
    ha[c].y = sH[m * MIDW + 4 * c + 2 * hi + 1];
  }
  // ---- layer 2: [16x32] @ [32x32] + b2, ReLU -> sH (reuse)
  {
    v2f b0[8], b1f[8];
    #pragma unroll
    for (int c = 0; c < 8; ++c) {
      b0[c]  = ldB(w2, MIDW, 4 * c, 0,  m, hi);
      b1f[c] = ldB(w2, MIDW, 4 * c, 16, m, hi);
    }
    float bias0 = bs2[m], bias1 = bs2[16 + m];
    v8f a0 = {}, a1 = {};
    #pragma unroll
    for (int c = 0; c < 8; ++c) {
      a0 = wmma4(ha[c], b0[c],  a0);
      a1 = wmma4(ha[c], b1f[c], a1);
    }
    #pragma unroll
    for (int r = 0; r < 8; ++r) {
      float v0 = a0[r] + bias0, v1 = a1[r] + bias1;
      sH[(r + 8 * hi) * MIDW + m]      = v0 > 0.f ? v0 : 0.f;
      sH[(r + 8 * hi) * MIDW + 16 + m] = v1 > 0.f ? v1 : 0.f;
    }
  }
  #pragma unroll
  for (int c = 0; c < 8; ++c) {
    ha[c].x = sH[m * MIDW + 4 * c + 2 * hi];
    ha[c].y = sH[m * MIDW + 4 * c + 2 * hi + 1];
  }
  // ---- layer 3: [16x32] @ [32 x NF*64] + b3 -> sR[16][NF*64]
  constexpr int NCOLS = NF * NCH * NCH;   // 64 or 192; NCOLS/16 is even
  #pragma unroll
  for (int j = 0; j < NCOLS / 16; j += 2) {
    v2f b0[8], b1f[8];
    #pragma unroll
    for (int c = 0; c < 8; ++c) {
      b0[c]  = ldB(w3, NCOLS, 4 * c, 16 * j,       m, hi);
      b1f[c] = ldB(w3, NCOLS, 4 * c, 16 * (j + 1), m, hi);
    }
    float bias0 = bs3[16 * j + m], bias1 = bs3[16 * (j + 1) + m];
    v8f a0 = {}, a1 = {};
    #pragma unroll
    for (int c = 0; c < 8; ++c) {
      a0 = wmma4(ha[c], b0[c],  a0);
      a1 = wmma4(ha[c], b1f[c], a1);
    }
    #pragma unroll
    for (int r = 0; r < 8; ++r) {
      sR[(r + 8 * hi) * NCOLS + 16 * j + m]       = a0[r] + bias0;
      sR[(r + 8 * hi) * NCOLS + 16 * (j + 1) + m] = a1[r] + bias1;
    }
  }
}

// Per-edge einsum: msg[o][p] += sum_{i,f} R[o,i,f] * (sum_q basis[p,q,f]*src[i,q])
// Two lanes per edge: `half` selects output channels o in [4*half, 4*half+4).
template<int DI, int DO, int NF>
__device__ __forceinline__ void edge_accum(
    const float* __restrict__ sRe,     // R for this edge: [NCH*NCH*NF]
    const float* __restrict__ basisE,  // basis for this edge: [PO][QI][NF]
    const float* __restrict__ srcF,    // gathered src feat: [NCH][QI]
    int half, float (&acc)[4][2 * DO + 1]) {
  constexpr int QI = 2 * DI + 1, PO = 2 * DO + 1;
  float s[NCH][QI];
  #pragma unroll
  for (int i = 0; i < NCH; ++i)
    #pragma unroll
    for (int q = 0; q < QI; ++q) s[i][q] = srcF[i * QI + q];
  float bas[PO][QI][NF];
  #pragma unroll
  for (int p = 0; p < PO; ++p)
    #pragma unroll
    for (int q = 0; q < QI; ++q)
      #pragma unroll
      for (int f = 0; f < NF; ++f) bas[p][q][f] = basisE[(p * QI + q) * NF + f];
  float T[NCH][NF][PO];
  #pragma unroll
  for (int i = 0; i < NCH; ++i)
    #pragma unroll
    for (int f = 0; f < NF; ++f)
      #pragma unroll
      for (int p = 0; p < PO; ++p) {
        float t = 0.f;
        #pragma unroll
        for (int q = 0; q < QI; ++q) t += bas[p][q][f] * s[i][q];
        T[i][f][p] = t;
      }
  #pragma unroll
  for (int ol = 0; ol < 4; ++ol) {
    int o = half * 4 + ol;
    const float* Ro = sRe + o * NCH * NF;
    #pragma unroll
    for (int p = 0; p < PO; ++p) {
      float t = 0.f;
      #pragma unroll
      for (int i = 0; i < NCH; ++i)
        #pragma unroll
        for (int f = 0; f < NF; ++f) t += Ro[i * NF + f] * T[i][f][p];
      acc[ol][p] += t;
    }
  }
}

__global__ void __launch_bounds__(WAVES_PER_BLK * 32)
se3_partial_kernel(KParams p) {
  __shared__ float sH[WAVES_PER_BLK][16 * MIDW];   //  2 KB/wave
  __shared__ float sR[WAVES_PER_BLK][16 * 192];    // 12 KB/wave (max NF=3)

  const int lane = threadIdx.x & 31;
  const int wv   = threadIdx.x >> 5;
  const int m    = lane & 15;   // edge-in-tile / matrix row
  const int hi   = lane >> 4;   // fragment half / o-channel half
  const int tile = blockIdx.x * WAVES_PER_BLK + wv;
  const int e0   = tile * 16;
  if (e0 >= E_TOT) return;      // wave-uniform (EXEC stays all-ones for WMMA)

  float* __restrict__ sHw = sH[wv];
  float* __restrict__ sRw = sR[wv];

  // ---- A-fragments of edge features ef = [w | radial], K padded 17 -> 20.
  // Chunks 0..3: two contiguous floats, 8B aligned -> single b64 load each.
  v2f xa[5];
  {
    const int er = e0 + m;
    #pragma unroll
    for (int c = 0; c < 4; ++c)
      xa[c] = *(const v2f*)(p.w + (size_t)er * EDIM + 4 * c + 2 * hi);
    float rad = p.radial[er];            // unconditional, in-bounds
    xa[4].x = hi ? 0.f : rad;            // k=16 for hi=0; k=18 (pad) for hi=1
    xa[4].y = 0.f;                       // k=17/19 (pad)
  }

  // this lane's edge for the einsum phase
  const int e = e0 + m;
  const int sidx = p.src_idx[e];
  const float* src0 = p.feat0 + (size_t)sidx * (NCH * 1);
  const float* src1 = p.feat1 + (size_t)sidx * (NCH * 3);

  float accD0[4][1] = {};
  float accD1[4][3] = {};

  // pair 0: (di=0, do=0), nf=1, basis [E,1,1,1]
  radial_mlp_tile<1>(p.w1[0], p.b1[0], p.w2[0], p.b2[0], p.w3[0], p.b3[0],
                     xa, sHw, sRw, m, hi);
  edge_accum<0, 0, 1>(sRw + m * 64, p.basis[0] + (size_t)e * 1, src0, hi, accD0);

  // pair 1: (di=0, do=1), nf=1, basis [E,3,1,1]
  radial_mlp_tile<1>(p.w1[1], p.b1[1], p.w2[1], p.b2[1], p.w3[1], p.b3[1],
                     xa, sHw, sRw, m, hi);
  edge_accum<0, 1, 1>(sRw + m * 64, p.basis[1] + (size_t)e * 3, src0, hi, accD1);

  // pair 2: (di=1, do=0), nf=1, basis [E,1,3,1]
  radial_mlp_tile<1>(p.w1[2], p.b1[2], p.w2[2], p.b2[2], p.w3[2], p.b3[2],
                     xa, sHw, sRw, m, hi);
  edge_accum<1, 0, 1>(sRw + m * 64, p.basis[2] + (size_t)e * 3, src1, hi, accD0);

  // pair 3: (di=1, do=1), nf=3, basis [E,3,3,3]
  radial_mlp_tile<3>(p.w1[3], p.b1[3], p.w2[3], p.b2[3], p.w3[3], p.b3[3],
                     xa, sHw, sRw, m, hi);
  edge_accum<1, 1, 3>(sRw + m * 192, p.basis[3] + (size_t)e * 27, src1, hi, accD1);

  // ---- write outputs: d_out = [ out0 (E*8) | out1 (E*24) ]
  #pragma unroll
  for (int ol = 0; ol < 4; ++ol) {
    int o = hi * 4 + ol;
    p.out[(size_t)e * NCH + o] = accD0[ol][0];
    #pragma unroll
    for (int q = 0; q < 3; ++q)
      p.out[(size_t)E_TOT * NCH + (size_t)e * (NCH * 3) + o * 3 + q] =
          accD1[ol][q];
  }
}

extern "C" void kernel_launch(void* const* d_in, const int* in_sizes, int n_in,
                              void* d_out, int out_size, void* d_ws,
                              size_t ws_size, hipStream_t stream) {
  (void)in_sizes; (void)n_in; (void)out_size; (void)d_ws; (void)ws_size;
  KParams p;
  p.feat0  = (const float*)d_in[0];
  p.feat1  = (const float*)d_in[1];
  p.w      = (const float*)d_in[2];
  p.radial = (const float*)d_in[3];
  for (int pr = 0; pr < 4; ++pr) {
    int b = 4 + pr * 7;
    p.basis[pr] = (const float*)d_in[b + 0];
    p.w1[pr]    = (const float*)d_in[b + 1];
    p.b1[pr]    = (const float*)d_in[b + 2];
    p.w2[pr]    = (const float*)d_in[b + 3];
    p.b2[pr]    = (const float*)d_in[b + 4];
    p.w3[pr]    = (const float*)d_in[b + 5];
    p.b3[pr]    = (const float*)d_in[b + 6];
  }
  p.src_idx = (const int*)d_in[32];
  p.out     = (float*)d_out;

  const int tiles  = E_TOT / 16;                                   // 10000
  const int blocks = (tiles + WAVES_PER_BLK - 1) / WAVES_PER_BLK;  // 2500
  hipLaunchKernelGGL(se3_partial_kernel, dim3(blocks), dim3(WAVES_PER_BLK * 32),
                     0, stream, p);
}